// DrugStructureEncoder_3934190043554
// MI455X (gfx1250) — compile-verified
//
#include <hip/hip_runtime.h>

typedef __attribute__((ext_vector_type(2))) float v2f;
typedef __attribute__((ext_vector_type(8))) float v8f;

#define DIM 128

// ---------------- degree / normalization ----------------
__global__ void k_init_deg(float* __restrict__ deg, int n) {
    int i = blockIdx.x * blockDim.x + threadIdx.x;
    if (i < n) deg[i] = 1.0f;  // self-loop
}

__global__ void k_count_deg(const int* __restrict__ dst, float* __restrict__ deg, int E) {
    int e = blockIdx.x * blockDim.x + threadIdx.x;
    if (e < E) unsafeAtomicAdd(&deg[dst[e]], 1.0f);
}

__global__ void k_rsqrt_inplace(float* __restrict__ deg, int n) {
    int i = blockIdx.x * blockDim.x + threadIdx.x;
    if (i < n) deg[i] = rsqrtf(deg[i]);  // deg >= 1 guaranteed
}

// ---------------- dense GEMM via fp32 WMMA ----------------
// out[nrows x 128] = A[nrows x 128] @ W[128 x 128], exact fp32 path.
// One wave -> 16 rows x 128 cols (8 column tiles of 16), K stepped by 4.
__global__ __launch_bounds__(128) void k_gemm_wmma(const float* __restrict__ A,
                                                   const float* __restrict__ W,
                                                   float* __restrict__ out,
                                                   int nrows) {
    const int wave = threadIdx.x >> 5;
    const int lane = threadIdx.x & 31;
    const int r0   = (blockIdx.x * 4 + wave) * 16;
    if (r0 >= nrows) return;               // wave-uniform: EXEC stays all-1s below

    const int m  = lane & 15;              // A row within tile / B,N column
    const int kh = (lane >> 4) * 2;        // K sub-offset: lanes 0-15 -> {0,1}, 16-31 -> {2,3}

    v8f acc[8] = {};                       // 8 tiles of 16x16 f32 accum

    const float* __restrict__ arow = A + (size_t)(r0 + m) * DIM;
    for (int k = 0; k < DIM; k += 4) {
        v2f a;
        a.x = arow[k + kh];
        a.y = arow[k + kh + 1];
        const float* __restrict__ wr0 = W + (size_t)(k + kh) * DIM;
        const float* __restrict__ wr1 = wr0 + DIM;
#pragma unroll
        for (int ct = 0; ct < 8; ++ct) {
            v2f b;
            b.x = wr0[ct * 16 + m];
            b.y = wr1[ct * 16 + m];
            acc[ct] = __builtin_amdgcn_wmma_f32_16x16x4_f32(
                false, a, false, b, (short)0, acc[ct], false, false);
        }
    }

    const int rr = r0 + (lane >> 4) * 8;   // D: VGPR j -> row rr+j, col = lane&15
#pragma unroll
    for (int ct = 0; ct < 8; ++ct) {
#pragma unroll
        for (int j = 0; j < 8; ++j) {
            out[(size_t)(rr + j) * DIM + ct * 16 + m] = acc[ct][j];
        }
    }
}

// ---------------- self-loop init: agg = t * dinv^2 (fully writes agg) ----------------
__global__ void k_selfloop(const float4* __restrict__ t, const float* __restrict__ dinv,
                           float4* __restrict__ agg, int n4) {   // n4 = N*32
    int i = blockIdx.x * blockDim.x + threadIdx.x;
    if (i >= n4) return;
    float di = dinv[i >> 5];
    float s  = di * di;
    float4 v = t[i];
    v.x *= s; v.y *= s; v.z *= s; v.w *= s;
    agg[i] = v;
}

// ---------------- per-edge gather/scale/scatter-add (one wave per edge) ----------------
__global__ __launch_bounds__(256) void k_edge_agg(const float* __restrict__ h,
                                                  const int* __restrict__ src,
                                                  const int* __restrict__ dst,
                                                  const float* __restrict__ dinv,
                                                  float* __restrict__ agg, int E) {
    int e = blockIdx.x * 8 + (threadIdx.x >> 5);
    if (e >= E) return;                    // wave-uniform
    int lane = threadIdx.x & 31;
    int s = src[e], d = dst[e];
    float nm = dinv[s] * dinv[d];
    float4 v = ((const float4*)(h + (size_t)s * DIM))[lane];
    float* ap = agg + (size_t)d * DIM + lane * 4;
    unsafeAtomicAdd(ap + 0, v.x * nm);
    unsafeAtomicAdd(ap + 1, v.y * nm);
    unsafeAtomicAdd(ap + 2, v.z * nm);
    unsafeAtomicAdd(ap + 3, v.w * nm);
}

// ---------------- bias (+ optional ReLU), float4-vectorized ----------------
__global__ void k_bias_act(const float4* __restrict__ a, const float4* __restrict__ b,
                           float4* __restrict__ o, int n4, int do_relu) {
    int i = blockIdx.x * blockDim.x + threadIdx.x;
    if (i >= n4) return;
    float4 v = a[i];
    float4 bb = b[i & 31];                 // 128 floats = 32 float4
    v.x += bb.x; v.y += bb.y; v.z += bb.z; v.w += bb.w;
    if (do_relu) {
        v.x = fmaxf(v.x, 0.f); v.y = fmaxf(v.y, 0.f);
        v.z = fmaxf(v.z, 0.f); v.w = fmaxf(v.w, 0.f);
    }
    o[i] = v;
}

// ---------------- pooling ----------------
__global__ void k_pool_init(float* __restrict__ pacc, float* __restrict__ pcnt, int G) {
    int i = blockIdx.x * blockDim.x + threadIdx.x;
    if (i < G * DIM) pacc[i] = 0.f;
    if (i < G) pcnt[i] = 0.f;
}

__global__ __launch_bounds__(256) void k_pool_acc(const float* __restrict__ h,
                                                  const int* __restrict__ batch,
                                                  float* __restrict__ pacc,
                                                  float* __restrict__ pcnt, int n) {
    int node = blockIdx.x * 8 + (threadIdx.x >> 5);
    if (node >= n) return;                 // wave-uniform
    int lane = threadIdx.x & 31;
    int g = batch[node];
    float4 v = ((const float4*)(h + (size_t)node * DIM))[lane];
    float* ap = pacc + (size_t)g * DIM + lane * 4;
    unsafeAtomicAdd(ap + 0, v.x);
    unsafeAtomicAdd(ap + 1, v.y);
    unsafeAtomicAdd(ap + 2, v.z);
    unsafeAtomicAdd(ap + 3, v.w);
    if (lane == 0) unsafeAtomicAdd(&pcnt[g], 1.0f);
}

__global__ void k_pool_final(const float* __restrict__ pacc, const float* __restrict__ pcnt,
                             float* __restrict__ out, int total) {
    int i = blockIdx.x * blockDim.x + threadIdx.x;
    if (i >= total) return;
    float c = pcnt[i >> 7];
    out[i] = pacc[i] / fmaxf(c, 1.0f);
}

extern "C" void kernel_launch(void* const* d_in, const int* in_sizes, int n_in,
                              void* d_out, int out_size, void* d_ws, size_t ws_size,
                              hipStream_t stream) {
    const float* x  = (const float*)d_in[0];
    const int* ei   = (const int*)d_in[1];
    const int* bat  = (const int*)d_in[2];
    const float* W1 = (const float*)d_in[3];
    const float* b1 = (const float*)d_in[4];
    const float* W2 = (const float*)d_in[5];
    const float* b2 = (const float*)d_in[6];
    float* out = (float*)d_out;

    const int N = in_sizes[0] / DIM;       // 50000
    const int E = in_sizes[1] / 2;         // 800000
    const int G = out_size / DIM;          // 512
    const int* src = ei;
    const int* dst = ei + E;

    // workspace layout (floats)
    float* deg  = (float*)d_ws;                          // N (becomes dinv in place)
    float* bufA = deg + (((size_t)N + 63) & ~(size_t)63);
    float* bufB = bufA + (size_t)N * DIM;
    float* pacc = bufB + (size_t)N * DIM;
    float* pcnt = pacc + (size_t)G * DIM;

    const int n4 = N * (DIM / 4);                        // float4 elements per feature buffer
    const int rowTiles = (N + 15) / 16;
    dim3 gemmGrid((rowTiles + 3) / 4);

    // 1) degree + dinv
    k_init_deg<<<(N + 255) / 256, 256, 0, stream>>>(deg, N);
    k_count_deg<<<(E + 255) / 256, 256, 0, stream>>>(dst, deg, E);
    k_rsqrt_inplace<<<(N + 255) / 256, 256, 0, stream>>>(deg, N);

    // 2) layer 1: t1 = x @ W1 -> bufA
    k_gemm_wmma<<<gemmGrid, 128, 0, stream>>>(x, W1, bufA, N);
    // agg1 (bufB) = t1*dinv^2 then scatter edges
    k_selfloop<<<(n4 + 255) / 256, 256, 0, stream>>>((const float4*)bufA, deg, (float4*)bufB, n4);
    k_edge_agg<<<(E + 7) / 8, 256, 0, stream>>>(bufA, src, dst, deg, bufB, E);
    // r1 (bufA) = relu(agg1 + b1)
    k_bias_act<<<(n4 + 255) / 256, 256, 0, stream>>>((const float4*)bufB, (const float4*)b1,
                                                     (float4*)bufA, n4, 1);

    // 3) layer 2: t2 = r1 @ W2 -> bufB
    k_gemm_wmma<<<gemmGrid, 128, 0, stream>>>(bufA, W2, bufB, N);
    k_selfloop<<<(n4 + 255) / 256, 256, 0, stream>>>((const float4*)bufB, deg, (float4*)bufA, n4);
    k_edge_agg<<<(E + 7) / 8, 256, 0, stream>>>(bufB, src, dst, deg, bufA, E);
    // h2 (bufA, in place) = agg2 + b2
    k_bias_act<<<(n4 + 255) / 256, 256, 0, stream>>>((const float4*)bufA, (const float4*)b2,
                                                     (float4*)bufA, n4, 0);

    // 4) global mean pool
    k_pool_init<<<(G * DIM + 255) / 256, 256, 0, stream>>>(pacc, pcnt, G);
    k_pool_acc<<<(N + 7) / 8, 256, 0, stream>>>(bufA, bat, pacc, pcnt, N);
    k_pool_final<<<(G * DIM + 255) / 256, 256, 0, stream>>>(pacc, pcnt, out, G * DIM);
}